// GCNConvLayer_3470333575821
// MI455X (gfx1250) — compile-verified
//
#include <hip/hip_runtime.h>

#define N_NODES 50000
#define N_EDGES 800000
#define DIM     128

typedef __attribute__((ext_vector_type(2))) float v2f;
typedef __attribute__((ext_vector_type(8))) float v8f;

// ---------------------------------------------------------------------------
// CSR construction
// ---------------------------------------------------------------------------
__global__ __launch_bounds__(256) void init_counts_kernel(int* counts) {
    int i = blockIdx.x * blockDim.x + threadIdx.x;
    if (i < N_NODES) counts[i] = 0;
}

__global__ __launch_bounds__(256) void count_edges_kernel(const int* __restrict__ dst,
                                                          int* __restrict__ counts) {
    int e = blockIdx.x * blockDim.x + threadIdx.x;
    if (e < N_EDGES) atomicAdd(&counts[dst[e]], 1);
}

// Single-block exclusive scan over counts -> offsets (and seeds fill cursors).
__global__ __launch_bounds__(1024) void scan_kernel(const int* __restrict__ counts,
                                                    int* __restrict__ offsets,
                                                    int* __restrict__ cursor) {
    __shared__ int sdata[1024];
    __shared__ int carry_s;
    const int tid = threadIdx.x;
    if (tid == 0) carry_s = 0;
    __syncthreads();
    for (int base = 0; base < N_NODES; base += 1024) {
        const int i = base + tid;
        const int v = (i < N_NODES) ? counts[i] : 0;
        sdata[tid] = v;
        __syncthreads();
        // Hillis-Steele inclusive scan
        for (int off = 1; off < 1024; off <<= 1) {
            int t = (tid >= off) ? sdata[tid - off] : 0;
            __syncthreads();
            sdata[tid] += t;
            __syncthreads();
        }
        const int carry = carry_s;
        if (i < N_NODES) {
            const int excl = carry + sdata[tid] - v;
            offsets[i] = excl;
            cursor[i]  = excl;
        }
        __syncthreads();
        if (tid == 1023) carry_s = carry + sdata[1023];
        __syncthreads();
    }
    if (tid == 0) offsets[N_NODES] = carry_s;
}

__global__ __launch_bounds__(256) void dinv_kernel(const int* __restrict__ counts,
                                                   float* __restrict__ dinv) {
    int i = blockIdx.x * blockDim.x + threadIdx.x;
    if (i < N_NODES) dinv[i] = rsqrtf((float)(counts[i] + 1)); // +1 self-loop
}

__global__ __launch_bounds__(256) void fill_csr_kernel(const int* __restrict__ src,
                                                       const int* __restrict__ dst,
                                                       int* __restrict__ cursor,
                                                       int* __restrict__ csr_col) {
    int e = blockIdx.x * blockDim.x + threadIdx.x;
    if (e < N_EDGES) {
        int pos = atomicAdd(&cursor[dst[e]], 1);
        csr_col[pos] = src[e];
    }
}

// Sort each node's adjacency segment -> deterministic float accumulation order.
__global__ __launch_bounds__(256) void sort_csr_kernel(const int* __restrict__ offsets,
                                                       int* __restrict__ csr_col) {
    int node = blockIdx.x * blockDim.x + threadIdx.x;
    if (node >= N_NODES) return;
    const int b = offsets[node], e = offsets[node + 1];
    for (int i = b + 1; i < e; ++i) {
        int key = csr_col[i];
        int j = i - 1;
        while (j >= b && csr_col[j] > key) { csr_col[j + 1] = csr_col[j]; --j; }
        csr_col[j + 1] = key;
    }
}

// ---------------------------------------------------------------------------
// Fused dual GEMM: h = x @ Wc, r = x @ Wr + res_b   (V_WMMA_F32_16X16X4_F32)
// Block = 256 threads = 8 waves. Block owns a 16-row M tile; wave w owns the
// 16-col N tile starting at 16*w. A tile staged once in LDS, reused by both
// accumulators (conv + residual).
// ---------------------------------------------------------------------------
__global__ __launch_bounds__(256) void gemm_dual_kernel(const float* __restrict__ x,
                                                        const float* __restrict__ Wc,
                                                        const float* __restrict__ Wr,
                                                        const float* __restrict__ res_b,
                                                        float* __restrict__ h,
                                                        float* __restrict__ r) {
    __shared__ float As[16 * DIM];
    const int m0  = blockIdx.x * 16;
    const int tid = threadIdx.x;

    // Cooperative load of the 16x128 A tile (each thread: 8 contiguous floats).
    {
        const int row = tid >> 4;           // 0..15
        const int col = (tid & 15) * 8;     // 0,8,...,120
        const float4* s4 = (const float4*)(x + (size_t)(m0 + row) * DIM + col);
        float4 p0 = s4[0], p1 = s4[1];
        float4* d4 = (float4*)(As + row * DIM + col);
        d4[0] = p0; d4[1] = p1;
    }
    __syncthreads();

    const int lane = tid & 31;
    const int wave = tid >> 5;              // 0..7
    const int nb   = wave * 16;             // N tile base
    const int ml   = lane & 15;
    const int hi   = lane >> 4;             // 0: K pair {0,1}; 1: K pair {2,3}

    v8f acc_h = {};
    v8f acc_r = {};
    for (int k = 0; k < DIM; k += 4) {
        const int ka = k + (hi << 1);       // lane-half K base
        v2f a, bc, br;
        // A fragment (16x4): lanes 0-15 -> K=k,k+1 ; lanes 16-31 -> K=k+2,k+3
        a.x = As[ml * DIM + ka];
        a.y = As[ml * DIM + ka + 1];
        // B fragments (4x16), row-major weights W[k][n]
        bc.x = Wc[(size_t)ka * DIM + nb + ml];
        bc.y = Wc[(size_t)(ka + 1) * DIM + nb + ml];
        br.x = Wr[(size_t)ka * DIM + nb + ml];
        br.y = Wr[(size_t)(ka + 1) * DIM + nb + ml];
        acc_h = __builtin_amdgcn_wmma_f32_16x16x4_f32(false, a, false, bc,
                                                      (short)0, acc_h, false, false);
        acc_r = __builtin_amdgcn_wmma_f32_16x16x4_f32(false, a, false, br,
                                                      (short)0, acc_r, false, false);
    }

    // C/D layout: VGPR i -> M = i + 8*hi ; N = lane&15
    const int col = nb + ml;
    const float rb = res_b[col];
#pragma unroll
    for (int i = 0; i < 8; ++i) {
        const int row = m0 + i + (hi << 3);
        h[(size_t)row * DIM + col] = acc_h[i];
        r[(size_t)row * DIM + col] = acc_r[i] + rb;
    }
}

// ---------------------------------------------------------------------------
// Aggregation: one wave32 per node, 4 features per lane (float4 = b128 loads).
// out[i] = relu( conv_b + dinv[i]^2*h[i] + sum_s dinv[s]*dinv[i]*h[s] + r[i] )
// ---------------------------------------------------------------------------
__global__ __launch_bounds__(256) void aggregate_kernel(const float* __restrict__ h,
                                                        const float* __restrict__ r,
                                                        const float* __restrict__ dinv,
                                                        const int* __restrict__ offsets,
                                                        const int* __restrict__ csr_col,
                                                        const float* __restrict__ conv_b,
                                                        float* __restrict__ out) {
    const int wave = threadIdx.x >> 5;
    const int lane = threadIdx.x & 31;
    const int node = blockIdx.x * 8 + wave;
    if (node >= N_NODES) return;
    const int c = lane * 4;

    float4 acc = *(const float4*)(conv_b + c);
    const float di = dinv[node];
    {
        const float ws = di * di; // self-loop norm
        const float4 hn = *(const float4*)(h + (size_t)node * DIM + c);
        acc.x += ws * hn.x; acc.y += ws * hn.y;
        acc.z += ws * hn.z; acc.w += ws * hn.w;
    }
    const int e0 = offsets[node];
    const int e1 = offsets[node + 1];
    for (int e = e0; e < e1; ++e) {
        const int s = csr_col[e];
        const float w = dinv[s] * di;
        const float4 hs = *(const float4*)(h + (size_t)s * DIM + c);
        acc.x += w * hs.x; acc.y += w * hs.y;
        acc.z += w * hs.z; acc.w += w * hs.w;
    }
    const float4 rr = *(const float4*)(r + (size_t)node * DIM + c);
    float4 o;
    o.x = fmaxf(acc.x + rr.x, 0.0f);
    o.y = fmaxf(acc.y + rr.y, 0.0f);
    o.z = fmaxf(acc.z + rr.z, 0.0f);
    o.w = fmaxf(acc.w + rr.w, 0.0f);
    *(float4*)(out + (size_t)node * DIM + c) = o;
}

// ---------------------------------------------------------------------------
extern "C" void kernel_launch(void* const* d_in, const int* in_sizes, int n_in,
                              void* d_out, int out_size, void* d_ws, size_t ws_size,
                              hipStream_t stream) {
    const float* x      = (const float*)d_in[0];
    const int*   edge   = (const int*)d_in[1];     // [2][E]: row0=src, row1=dst
    const float* conv_w = (const float*)d_in[2];   // [2][D][D]
    const float* conv_b = (const float*)d_in[3];   // [2][D]
    const float* res_w  = (const float*)d_in[4];   // [D][D]
    const float* res_b  = (const float*)d_in[5];   // [D]
    float*       out    = (float*)d_out;

    const int* e_src = edge;
    const int* e_dst = edge + N_EDGES;

    // Workspace carving (256B aligned)
    char* base = (char*)d_ws;
    size_t off = 0;
    auto alloc = [&](size_t bytes) -> char* {
        char* p = base + off;
        off += (bytes + 255) & ~(size_t)255;
        return p;
    };
    int*   counts  = (int*)  alloc((size_t)N_NODES * sizeof(int));
    int*   offs    = (int*)  alloc((size_t)(N_NODES + 1) * sizeof(int));
    int*   cursor  = (int*)  alloc((size_t)N_NODES * sizeof(int));
    int*   csr     = (int*)  alloc((size_t)N_EDGES * sizeof(int));
    float* dinv    = (float*)alloc((size_t)N_NODES * sizeof(float));
    float* hbuf    = (float*)alloc((size_t)N_NODES * DIM * sizeof(float));
    float* rbuf    = (float*)alloc((size_t)N_NODES * DIM * sizeof(float));
    (void)ws_size; (void)in_sizes; (void)n_in; (void)out_size;

    const int gN = (N_NODES + 255) / 256;   // 196
    const int gE = (N_EDGES + 255) / 256;   // 3125
    const int gM = N_NODES / 16;            // 3125 (50000 % 16 == 0)
    const int gA = (N_NODES + 7) / 8;       // 6250

    // CSR build (reused by both layers)
    init_counts_kernel<<<gN, 256, 0, stream>>>(counts);
    count_edges_kernel<<<gE, 256, 0, stream>>>(e_dst, counts);
    scan_kernel<<<1, 1024, 0, stream>>>(counts, offs, cursor);
    dinv_kernel<<<gN, 256, 0, stream>>>(counts, dinv);
    fill_csr_kernel<<<gE, 256, 0, stream>>>(e_src, e_dst, cursor, csr);
    sort_csr_kernel<<<gN, 256, 0, stream>>>(offs, csr);

    // Layer 0: reads x (input), writes out
    gemm_dual_kernel<<<gM, 256, 0, stream>>>(x, conv_w, res_w, res_b, hbuf, rbuf);
    aggregate_kernel<<<gA, 256, 0, stream>>>(hbuf, rbuf, dinv, offs, csr, conv_b, out);

    // Layer 1: reads out, writes out (GEMM consumes x before aggregate overwrites)
    gemm_dual_kernel<<<gM, 256, 0, stream>>>(out, conv_w + DIM * DIM, res_w, res_b, hbuf, rbuf);
    aggregate_kernel<<<gA, 256, 0, stream>>>(hbuf, rbuf, dinv, offs, csr, conv_b + DIM, out);
}